// NeuralGraphOutput_32504312496731
// MI455X (gfx1250) — compile-verified
//
#include <hip/hip_runtime.h>

// CDNA5 / gfx1250, wave32. f32 WMMA path: V_WMMA_F32_16X16X4_F32.
typedef float v2f __attribute__((ext_vector_type(2)));
typedef float v8f __attribute__((ext_vector_type(8)));

#define NBATCH 2048
#define NA     256   // atoms per batch
#define ND     8     // neighbor slots
#define FAtm   64    // atom features
#define FBnd   16    // bond features
#define FPout  256   // fingerprint width
#define KDIM   80    // FAtm + FBnd
#define XSTR   84    // padded K: 80 feats + mask col + 3 zeros (also bank-friendly stride)
#define KSTEPS 21    // XSTR / 4

__global__ __launch_bounds__(256)
void ngf_wmma_kernel(const float* __restrict__ atoms,
                     const float* __restrict__ bonds,
                     const int*   __restrict__ edges,
                     const float* __restrict__ W,      // [80][256] row-major
                     const float* __restrict__ bias,   // [256]
                     float* __restrict__ out)          // [2048][256]
{
    extern __shared__ float lds_x[];          // [NA][XSTR] = 84 KB
    const int b   = blockIdx.x;
    const int tid = threadIdx.x;

    // ---------------- Phase 1: stage X' = mask * [atoms | sum_d bonds | 1 | 0 0 0] ----------------
    {
        const int a = tid;                              // one atom per thread
        const size_t ab = (size_t)b * NA + a;

        // degree mask from edge list (-1 = padded slot)
        const int* e = edges + ab * ND;
        int deg = 0;
        #pragma unroll
        for (int d = 0; d < ND; ++d) deg += (e[d] != -1) ? 1 : 0;
        const float maskf = (deg != 0) ? 1.0f : 0.0f;

        float* xrow = lds_x + a * XSTR;

        // atom features (64 floats, vectorized)
        const float4* arow = (const float4*)(atoms + ab * FAtm);
        #pragma unroll
        for (int f4 = 0; f4 < FAtm / 4; ++f4) {
            float4 v = arow[f4];
            xrow[f4 * 4 + 0] = maskf * v.x;
            xrow[f4 * 4 + 1] = maskf * v.y;
            xrow[f4 * 4 + 2] = maskf * v.z;
            xrow[f4 * 4 + 3] = maskf * v.w;
        }

        // summed bond features (8 x 16 floats -> 16 sums)
        const float4* brow = (const float4*)(bonds + ab * (size_t)(ND * FBnd));
        float4 s[FBnd / 4] = {};
        #pragma unroll
        for (int d = 0; d < ND; ++d) {
            #pragma unroll
            for (int f4 = 0; f4 < FBnd / 4; ++f4) {
                float4 v = brow[d * (FBnd / 4) + f4];
                s[f4].x += v.x; s[f4].y += v.y; s[f4].z += v.z; s[f4].w += v.w;
            }
        }
        #pragma unroll
        for (int f4 = 0; f4 < FBnd / 4; ++f4) {
            xrow[FAtm + f4 * 4 + 0] = maskf * s[f4].x;
            xrow[FAtm + f4 * 4 + 1] = maskf * s[f4].y;
            xrow[FAtm + f4 * 4 + 2] = maskf * s[f4].z;
            xrow[FAtm + f4 * 4 + 3] = maskf * s[f4].w;
        }
        xrow[80] = maskf;                 // multiplies bias row of W'
        xrow[81] = 0.0f;
        xrow[82] = 0.0f;
        xrow[83] = 0.0f;
    }
    __syncthreads();

    // ---------------- Phase 2: [256 x 84] @ [84 x 256] via WMMA, ReLU, row-reduce ----------------
    const int wave = tid >> 5;
    const int lane = tid & 31;
    const int hi   = lane >> 4;          // lane half: picks K+2/K+3 rows of A/B fragments
    const int lsub = lane & 15;
    const int n0   = wave * 32;          // this wave owns cols [n0, n0+31] as two 16-wide tiles

    // Preload B fragments (held in VGPRs across the whole M loop).
    // B 4x16 f32 layout: VGPR0 = row K (lanes 0-15) / K+2 (lanes 16-31); VGPR1 = K+1 / K+3.
    v2f bf0[KSTEPS], bf1[KSTEPS];
    #pragma unroll
    for (int ks = 0; ks < 20; ++ks) {
        const int k0 = ks * 4 + 2 * hi;
        bf0[ks].x = W[(size_t)(k0    ) * FPout + n0 + lsub];
        bf0[ks].y = W[(size_t)(k0 + 1) * FPout + n0 + lsub];
        bf1[ks].x = W[(size_t)(k0    ) * FPout + n0 + 16 + lsub];
        bf1[ks].y = W[(size_t)(k0 + 1) * FPout + n0 + 16 + lsub];
    }
    // Last K-step rows are [bias, 0, 0, 0] (paired with the mask column of X').
    bf0[20].x = (hi == 0) ? bias[n0 + lsub]      : 0.0f;
    bf0[20].y = 0.0f;
    bf1[20].x = (hi == 0) ? bias[n0 + 16 + lsub] : 0.0f;
    bf1[20].y = 0.0f;

    v8f run0 = {};
    v8f run1 = {};

    for (int mt = 0; mt < NA / 16; ++mt) {
        // A 16x4 f32 layout: lane holds row M=lane&15; lanes 0-15 get {K,K+1}, 16-31 get {K+2,K+3}
        const float* xr = lds_x + (mt * 16 + lsub) * XSTR + 2 * hi;
        v8f acc0 = {};
        v8f acc1 = {};
        #pragma unroll
        for (int ks = 0; ks < KSTEPS; ++ks) {
            v2f a = *(const v2f*)(xr + ks * 4);   // 8-byte aligned LDS load
            acc0 = __builtin_amdgcn_wmma_f32_16x16x4_f32(
                false, a, false, bf0[ks], (short)0, acc0, false, false);
            acc1 = __builtin_amdgcn_wmma_f32_16x16x4_f32(
                false, a, false, bf1[ks], (short)0, acc1, false, false);
        }
        // ReLU then accumulate the 16-row stripe into the running column sums
        #pragma unroll
        for (int r = 0; r < 8; ++r) {
            run0[r] += fmaxf(acc0[r], 0.0f);
            run1[r] += fmaxf(acc1[r], 0.0f);
        }
    }

    // Per-lane: sum 8 D VGPRs (rows r or r+8 of every M tile), then fold lane halves.
    float s0 = 0.0f, s1 = 0.0f;
    #pragma unroll
    for (int r = 0; r < 8; ++r) { s0 += run0[r]; s1 += run1[r]; }
    s0 += __shfl_xor(s0, 16, 32);
    s1 += __shfl_xor(s1, 16, 32);

    if (hi == 0) {
        out[(size_t)b * FPout + n0 + lsub]      = s0;
        out[(size_t)b * FPout + n0 + 16 + lsub] = s1;
    }
}

extern "C" void kernel_launch(void* const* d_in, const int* in_sizes, int n_in,
                              void* d_out, int out_size, void* d_ws, size_t ws_size,
                              hipStream_t stream) {
    const float* atoms = (const float*)d_in[0];
    const float* bonds = (const float*)d_in[1];
    const int*   edges = (const int*)d_in[2];
    const float* W     = (const float*)d_in[3];
    const float* bias  = (const float*)d_in[4];
    float* out = (float*)d_out;

    const size_t smem = (size_t)NA * XSTR * sizeof(float);   // 86016 B dynamic LDS
    ngf_wmma_kernel<<<NBATCH, 256, smem, stream>>>(atoms, bonds, edges, W, bias, out);
}